// CausalDepthwiseConv1d_31215822307431
// MI455X (gfx1250) — compile-verified
//
#include <hip/hip_runtime.h>
#include <hip/hip_bf16.h>
#include <cstdint>

// Problem constants (match reference)
#define B_  4
#define S_  8192
#define H_  2048
#define K_  4

// Tiling
#define THREADS 128           // lanes per block; each owns 4 channels -> 512 ch/block
#define HB      (THREADS * 4) // 512 channels per block
#define RBUF    16            // LDS ring rows (power of two)
#define DEPTH   12            // async pipeline depth (<= RBUF - 4)
#define SEG     128           // timesteps per block segment (multiple of DEPTH not required)

typedef float v4f __attribute__((ext_vector_type(4)));

// Per-lane async copy: 16B global -> LDS, tracked by ASYNCcnt (CDNA5 path).
// GVS mode: lds_off (VGPR, wave-relative LDS byte addr), voff (VGPR, 32-bit
// unsigned byte offset), sbase (SGPR pair, 64-bit base).
__device__ __forceinline__ void async_ld_b128(uint32_t lds_off, uint32_t voff,
                                              const float* sbase) {
  asm volatile("global_load_async_to_lds_b128 %0, %1, %2"
               :: "v"(lds_off), "v"(voff), "s"(sbase)
               : "memory");
}
__device__ __forceinline__ void wait_async_depth() {
  // Keep literal in sync with DEPTH (=12)
  asm volatile("s_wait_asynccnt 12" ::: "memory");
}
__device__ __forceinline__ void wait_async_zero() {
  asm volatile("s_wait_asynccnt 0" ::: "memory");
}

__device__ __forceinline__ uint32_t lds_byte_addr(const void* p) {
  // flat shared addr low 32 bits == wave-relative LDS byte offset
  return (uint32_t)(uintptr_t)p;
}

__device__ __forceinline__ v4f conv_silu(v4f xm3, v4f xm2, v4f xm1, v4f xt,
                                         v4f w0, v4f w1, v4f w2, v4f w3) {
  // tap i multiplies x[t-3+i]; lane's 4 channels use weight rows w0..w3
  v4f y;
  y.x = w0.x * xm3.x + w0.y * xm2.x + w0.z * xm1.x + w0.w * xt.x;
  y.y = w1.x * xm3.y + w1.y * xm2.y + w1.z * xm1.y + w1.w * xt.y;
  y.z = w2.x * xm3.z + w2.y * xm2.z + w2.z * xm1.z + w2.w * xt.z;
  y.w = w3.x * xm3.w + w3.y * xm2.w + w3.z * xm1.w + w3.w * xt.w;
  v4f s;
  s.x = 1.f / (1.f + __expf(-y.x));
  s.y = 1.f / (1.f + __expf(-y.y));
  s.z = 1.f / (1.f + __expf(-y.z));
  s.w = 1.f / (1.f + __expf(-y.w));
  return y * s;
}

__global__ __launch_bounds__(THREADS)
void CausalDepthwiseConv1d_31215822307431_kernel(const float* __restrict__ x,
                                                 const float* __restrict__ w,
                                                 float* __restrict__ out) {
  __shared__ v4f sx[RBUF * THREADS];   // 16 rows * 128 lanes * 16B = 32 KB

  const int tid   = threadIdx.x;
  const int b     = blockIdx.z;
  const int hbase = blockIdx.y * HB;
  const int t0    = blockIdx.x * SEG;
  const int h0    = hbase + tid * 4;

  const float* xb = x   + (size_t)b * S_ * H_;
  float*       ob = out + (size_t)b * S_ * H_;

  // Per-thread weights: rows h0..h0+3 of [H,K=4] are contiguous float4s.
  const v4f* w4 = (const v4f*)w;
  const v4f wq0 = w4[h0 + 0];
  const v4f wq1 = w4[h0 + 1];
  const v4f wq2 = w4[h0 + 2];
  const v4f wq3 = w4[h0 + 3];

  const uint32_t lds_col = lds_byte_addr(&sx[tid]);

  // ---- Prologue: stage rows t0-3 .. t0+DEPTH-1 (zeros for t<0, causal pad) ----
#pragma unroll
  for (int r = -3; r < DEPTH; ++r) {
    const int t = t0 + r;
    const uint32_t slot = (uint32_t)(t & (RBUF - 1));
    if (t < 0) {                       // uniform branch (only t0==0 segment)
      v4f z = {0.f, 0.f, 0.f, 0.f};
      sx[slot * THREADS + tid] = z;
    } else {
      async_ld_b128(lds_col + slot * (THREADS * 16u),
                    ((uint32_t)t * H_ + (uint32_t)h0) * 4u, xb);
    }
  }
  wait_async_depth();                  // oldest 3 rows complete

  v4f xm3 = sx[((t0 - 3) & (RBUF - 1)) * THREADS + tid];
  v4f xm2 = sx[((t0 - 2) & (RBUF - 1)) * THREADS + tid];
  v4f xm1 = sx[((t0 - 1) & (RBUF - 1)) * THREADS + tid];

  const int tend = t0 + SEG;

  // ---- Main pipelined loop: issue row t+DEPTH, wait<=DEPTH, compute row t ----
#pragma unroll 4
  for (int t = t0; t < tend - DEPTH; ++t) {
    const int tp = t + DEPTH;
    async_ld_b128(lds_col + (uint32_t)(tp & (RBUF - 1)) * (THREADS * 16u),
                  ((uint32_t)tp * H_ + (uint32_t)h0) * 4u, xb);
    wait_async_depth();
    v4f xt = sx[(t & (RBUF - 1)) * THREADS + tid];
    v4f y  = conv_silu(xm3, xm2, xm1, xt, wq0, wq1, wq2, wq3);
    __builtin_nontemporal_store(y, (v4f*)(ob + (size_t)t * H_ + h0));
    xm3 = xm2; xm2 = xm1; xm1 = xt;
  }

  // ---- Tail: drain pipeline, compute remaining DEPTH rows ----
  wait_async_zero();
#pragma unroll 4
  for (int t = tend - DEPTH; t < tend; ++t) {
    v4f xt = sx[(t & (RBUF - 1)) * THREADS + tid];
    v4f y  = conv_silu(xm3, xm2, xm1, xt, wq0, wq1, wq2, wq3);
    __builtin_nontemporal_store(y, (v4f*)(ob + (size_t)t * H_ + h0));
    xm3 = xm2; xm2 = xm1; xm1 = xt;
  }
}

extern "C" void kernel_launch(void* const* d_in, const int* in_sizes, int n_in,
                              void* d_out, int out_size, void* d_ws, size_t ws_size,
                              hipStream_t stream) {
  (void)in_sizes; (void)n_in; (void)d_ws; (void)ws_size; (void)out_size;
  const float* x = (const float*)d_in[0];   // (B,S,H) fp32
  const float* w = (const float*)d_in[1];   // (H,K=4) fp32
  float* out = (float*)d_out;               // (B,S,H) fp32

  dim3 grid(S_ / SEG, H_ / HB, B_);         // (64, 4, 4) = 1024 blocks
  dim3 block(THREADS);                      // 128 threads = 4 waves (wave32)
  CausalDepthwiseConv1d_31215822307431_kernel<<<grid, block, 0, stream>>>(x, w, out);
}